// EMA_5995774345648
// MI455X (gfx1250) — compile-verified
//
#include <hip/hip_runtime.h>
#include <math.h>

// EMA fused kernel for MI455X (gfx1250).
// One workgroup per (batch, group) = 512 workgroups; each owns a 16ch x 64x64
// fp32 tile (256KB) resident in LDS (320KB/WGP). Single HBM read via TDM,
// single HBM write. All matmuls (1x1 convs + 3x3 conv) on V_WMMA_F32_16X16X4_F32.

constexpr int kCG      = 16;    // channels per group
constexpr int kHWX     = 64;    // spatial edge
constexpr int kHW      = 4096;  // 64*64
constexpr int kThreads = 512;   // 16 wave32
constexpr int kWaves   = 16;
constexpr int kBG      = 512;   // B*G workgroups

typedef float v8f  __attribute__((ext_vector_type(8)));
typedef float v2f  __attribute__((ext_vector_type(2)));
typedef unsigned int u32x4 __attribute__((ext_vector_type(4)));
typedef int   i32x4 __attribute__((ext_vector_type(4)));
typedef int   i32x8 __attribute__((ext_vector_type(8)));

struct SMem {
  float xg[kCG * kHW];            // 256 KB  input tile (TDM target, offset 0)
  float ybuf[kCG * 128];          // concat(x_h, x_w); conv1 relu output in-place
  float ah[kCG * kHWX];           // sigmoid(convh(y_h))
  float aw[kCG * kHWX];           // sigmoid(convw(y_w))
  float w3s[9 * kCG * kCG];       // conv3 weights re-laid as [tap][o][i]
  float w1s[kCG * kCG], whs[kCG * kCG], wws[kCG * kCG];
  float b1s[kCG], bhs[kCG], bws[kCG], b3s[kCG], gnw[kCG], gnb[kCG];
  float ss[kHW];                  // s2 + s3 accumulator
  float wpm[kWaves * kCG], wps[kWaves * kCG], wpv[kWaves * kCG]; // per-wave softmax partials
  float red[kWaves], red2[kWaves];
  float m2[kCG], se2[kCG], V2[kCG], m3[kCG], se3[kCG], V3[kCG];
  float scal[4];                  // mu, rsig, softmax max, softmax sum
};

// D = A(16x4 f32) * B(4x16 f32) + C, fp32 accumulate on the matrix pipe.
__device__ __forceinline__ v8f wmma4(v2f a, v2f b, v8f c) {
  return __builtin_amdgcn_wmma_f32_16x16x4_f32(false, a, false, b, (short)0, c,
                                               false, false);
}

// 3x3 conv, one 16(out-ch) x 16(position) output tile at row h, cols w0..w0+15.
// 9 taps x (K=16 in-ch as 4 chained K=4 WMMAs) = 36 WMMAs, zero-padded borders.
__device__ __forceinline__ v8f conv3_tile(const SMem& sm, int h, int w0,
                                          int lr, int half) {
  v8f acc = {0.f,0.f,0.f,0.f,0.f,0.f,0.f,0.f};
#pragma unroll
  for (int tap = 0; tap < 9; ++tap) {
    const int dh = tap / 3 - 1, dw = tap % 3 - 1;
    const int hh = h + dh;
    const int cc = w0 + lr + dw;
    const bool inb = (hh >= 0) & (hh < kHWX) & (cc >= 0) & (cc < kHWX);
    const int xoff = hh * kHWX + cc;
    const float* Wt = sm.w3s + tap * 256;
#pragma unroll
    for (int k0 = 0; k0 < 16; k0 += 4) {
      const int ka = k0 + 2 * half;            // A/B K-pair owned by this half
      v2f a, b;
      a.x = Wt[lr * 16 + ka];                  // A: M=lr, K=ka / ka+1
      a.y = Wt[lr * 16 + ka + 1];
      b.x = inb ? sm.xg[ka * kHW + xoff] : 0.f;        // B: K=ka, N=pos
      b.y = inb ? sm.xg[(ka + 1) * kHW + xoff] : 0.f;
      acc = wmma4(a, b, acc);
    }
  }
  return acc;
}

__global__ void __launch_bounds__(kThreads, 1)
ema_fused_kernel(const float* __restrict__ x,
                 const float* __restrict__ w1, const float* __restrict__ b1,
                 const float* __restrict__ wh, const float* __restrict__ bh,
                 const float* __restrict__ ww, const float* __restrict__ bw,
                 const float* __restrict__ w3, const float* __restrict__ b3,
                 const float* __restrict__ gw, const float* __restrict__ gb,
                 float* __restrict__ out) {
  extern __shared__ char smem_raw[];
  SMem& sm = *reinterpret_cast<SMem*>(smem_raw);

  const int tid  = threadIdx.x;
  const int lane = tid & 31;
  const int wave = tid >> 5;
  const int half = (lane >> 4) & 1;
  const int lr   = lane & 15;
  const int g    = blockIdx.x;
  const float* xgrp = x + (size_t)g * (kCG * kHW);
  float* ogrp = out + (size_t)g * (kCG * kHW);

  // ---- Phase 0: stage weights (overlapped with TDM tile load below) --------
  for (int i = tid; i < 256; i += kThreads) {
    sm.w1s[i] = w1[i]; sm.whs[i] = wh[i]; sm.wws[i] = ww[i];
  }
  for (int i = tid; i < 9 * 256; i += kThreads) {
    const int o = i / 144, rem = i % 144, ic = rem / 9, tap = rem % 9;
    sm.w3s[tap * 256 + o * 16 + ic] = w3[i];   // (o,i,kh,kw) -> [tap][o][i]
  }
  if (tid < kCG) {
    sm.b1s[tid] = b1[tid]; sm.bhs[tid] = bh[tid]; sm.bws[tid] = bw[tid];
    sm.b3s[tid] = b3[tid]; sm.gnw[tid] = gw[tid]; sm.gnb[tid] = gb[tid];
  }

  // ---- Phase 1: 256KB tile -> LDS via Tensor Data Mover -------------------
#if __has_builtin(__builtin_amdgcn_tensor_load_to_lds) && \
    __has_builtin(__builtin_amdgcn_s_wait_tensorcnt)
  if (wave == 0) {                 // one TDM op per workgroup
    const unsigned long long ga = (unsigned long long)xgrp;
    u32x4 g0;                      // D# group 0
    g0[0] = 1u;                    // count=1, user descriptor, no gather
    g0[1] = 0u;                    // lds_addr: sm.xg is at LDS offset 0
    g0[2] = (unsigned)(ga & 0xffffffffull);
    g0[3] = (unsigned)((ga >> 32) & 0x1ffffffull) | (2u << 30); // addr[56:32] | type=2
    i32x8 g1;                      // D# group 1: 2D tensor 4096 x 16, 4B elems
    g1[0] = 0x20000;               // data_size=2 (4 bytes), no multicast/pad
    g1[1] = (int)(4096u << 16);    // tensor_dim0[15:0] in bits [63:48]
    g1[2] = (int)(16u << 16);      // tensor_dim0 hi=0 | tensor_dim1=16
    g1[3] = (int)(4096u << 16);    // tensor_dim1 hi=0 | tile_dim0=4096
    g1[4] = 16;                    // tile_dim1=16, tile_dim2=0
    g1[5] = 4096;                  // tensor_dim0_stride lo32
    g1[6] = 0;                     // stride0 hi | stride1[15:0] (65536 & 0xffff)
    g1[7] = 1;                     // stride1[47:16] = 65536 >> 16
    i32x4 z4 = {0, 0, 0, 0};
#if defined(__clang_major__) && (__clang_major__ >= 23)
    i32x8 z8 = {0, 0, 0, 0, 0, 0, 0, 0};
    __builtin_amdgcn_tensor_load_to_lds(g0, g1, z4, z4, z8, 0);
#else
    __builtin_amdgcn_tensor_load_to_lds(g0, g1, z4, z4, 0);
#endif
    __builtin_amdgcn_s_wait_tensorcnt(0);
  }
#else
  for (int i = tid; i < kCG * kHW / 4; i += kThreads)
    reinterpret_cast<float4*>(sm.xg)[i] =
        reinterpret_cast<const float4*>(xgrp)[i];
#endif
  __syncthreads();

  // ---- Phase 2: row / column means -> ybuf = concat(x_h, x_w) -------------
  for (int idx = tid; idx < kCG * kHWX; idx += kThreads) {
    const int c = idx >> 6, j = idx & 63;
    float sh = 0.f, sw = 0.f;
    for (int k = 0; k < kHWX; ++k) {
      sh += sm.xg[c * kHW + j * kHWX + k];   // mean over W -> x_h[c][j]
      sw += sm.xg[c * kHW + k * kHWX + j];   // mean over H -> x_w[c][j]
    }
    sm.ybuf[c * 128 + j]      = sh * (1.f / 64.f);
    sm.ybuf[c * 128 + 64 + j] = sw * (1.f / 64.f);
  }
  __syncthreads();

  // ---- Phase 3: conv1 (16x16 @ 16x128) via WMMA, ReLU, in place -----------
  if (wave < 8) {
    const int n0 = wave * 16;
    v8f acc = {0.f,0.f,0.f,0.f,0.f,0.f,0.f,0.f};
#pragma unroll
    for (int k0 = 0; k0 < 16; k0 += 4) {
      const int ka = k0 + 2 * half;
      v2f a, b;
      a.x = sm.w1s[lr * 16 + ka];  a.y = sm.w1s[lr * 16 + ka + 1];
      b.x = sm.ybuf[ka * 128 + n0 + lr];
      b.y = sm.ybuf[(ka + 1) * 128 + n0 + lr];
      acc = wmma4(a, b, acc);
    }
#pragma unroll
    for (int r = 0; r < 8; ++r) {
      const int row = r + 8 * half;
      const float v = acc[r] + sm.b1s[row];
      sm.ybuf[row * 128 + n0 + lr] = v > 0.f ? v : 0.f;
    }
  }
  __syncthreads();

  // ---- Phase 4: convh / convw via WMMA, sigmoid -> ah, aw -----------------
  if (wave < 8) {
    const bool ish = wave < 4;
    const int n0 = (ish ? wave : wave - 4) * 16;
    const float* Wm = ish ? sm.whs : sm.wws;
    const float* bm = ish ? sm.bhs : sm.bws;
    const int srcCol = (ish ? 0 : 64) + n0 + lr;
    v8f acc = {0.f,0.f,0.f,0.f,0.f,0.f,0.f,0.f};
#pragma unroll
    for (int k0 = 0; k0 < 16; k0 += 4) {
      const int ka = k0 + 2 * half;
      v2f a, b;
      a.x = Wm[lr * 16 + ka];  a.y = Wm[lr * 16 + ka + 1];
      b.x = sm.ybuf[ka * 128 + srcCol];
      b.y = sm.ybuf[(ka + 1) * 128 + srcCol];
      acc = wmma4(a, b, acc);
    }
    float* dst = ish ? sm.ah : sm.aw;
#pragma unroll
    for (int r = 0; r < 8; ++r) {
      const int row = r + 8 * half;
      const float v = acc[r] + bm[row];
      dst[row * kHWX + n0 + lr] = 1.f / (1.f + expf(-v));
    }
  }
  __syncthreads();

  // ---- Phase 5: group-norm stats over x1 = xg * a_h * a_w -----------------
  {
    float s = 0.f, q = 0.f;
    for (int i = tid; i < kCG * kHW; i += kThreads) {
      const int c = i >> 12, l = i & 4095, h = l >> 6, wc = l & 63;
      const float v = sm.xg[i] * sm.ah[c * kHWX + h] * sm.aw[c * kHWX + wc];
      s += v;  q += v * v;
    }
    for (int off = 16; off; off >>= 1) {
      s += __shfl_xor(s, off);  q += __shfl_xor(q, off);
    }
    if (lane == 0) { sm.red[wave] = s; sm.red2[wave] = q; }
  }
  __syncthreads();
  if (tid == 0) {
    float S = 0.f, Q = 0.f;
    for (int i = 0; i < kWaves; ++i) { S += sm.red[i]; Q += sm.red2[i]; }
    const float mu = S * (1.f / 65536.f);
    const float var = Q * (1.f / 65536.f) - mu * mu;
    sm.scal[0] = mu;
    sm.scal[1] = rsqrtf(var + 1e-5f);
  }
  __syncthreads();

  // ---- Phase 6: per-channel online softmax stats of x2 (wave c = channel c)
  {
    const int c = wave;
    const float mu = sm.scal[0], rs = sm.scal[1];
    const float cgw = sm.gnw[c], cgb = sm.gnb[c];
    float m = -__builtin_inff(), s = 0.f, sv = 0.f;
    for (int l = lane; l < kHW; l += 32) {
      const int h = l >> 6, wc = l & 63;
      const float x1v = sm.xg[c * kHW + l] * sm.ah[c * kHWX + h] * sm.aw[c * kHWX + wc];
      const float v = (x1v - mu) * rs * cgw + cgb;
      const float mn = fmaxf(m, v);
      s = s * expf(m - mn) + expf(v - mn);
      m = mn;  sv += v;
    }
    for (int off = 16; off; off >>= 1) {
      const float mo = __shfl_xor(m, off), so = __shfl_xor(s, off),
                  svo = __shfl_xor(sv, off);
      const float mn = fmaxf(m, mo);
      s = s * expf(m - mn) + so * expf(mo - mn);
      m = mn;  sv += svo;
    }
    if (lane == 0) { sm.m2[c] = m; sm.se2[c] = s; sm.V2[c] = sv * (1.f / 4096.f); }
  }
  __syncthreads();

  // ---- Phase 7: conv3 pass #1 (WMMA) + per-channel online stats of x3 -----
  {
    float m3r[8], s3r[8], v3r[8];
#pragma unroll
    for (int r = 0; r < 8; ++r) { m3r[r] = -__builtin_inff(); s3r[r] = 0.f; v3r[r] = 0.f; }
    for (int t = wave; t < 256; t += kWaves) {
      const int h = t >> 2, w0 = (t & 3) << 4;
      v8f acc = conv3_tile(sm, h, w0, lr, half);
#pragma unroll
      for (int r = 0; r < 8; ++r) {
        const float v = fmaxf(acc[r] + sm.b3s[r + 8 * half], 0.f);
        float m = v, s = 1.f, sv = v;
#pragma unroll
        for (int off = 1; off < 16; off <<= 1) {   // intra-half (same channel)
          const float mo = __shfl_xor(m, off), so = __shfl_xor(s, off),
                      svo = __shfl_xor(sv, off);
          const float mn = fmaxf(m, mo);
          s = s * expf(m - mn) + so * expf(mo - mn);
          m = mn;  sv += svo;
        }
        const float mn = fmaxf(m3r[r], m);
        s3r[r] = s3r[r] * expf(m3r[r] - mn) + s * expf(m - mn);
        m3r[r] = mn;  v3r[r] += sv;
      }
    }
    if (lr == 0) {                 // lane 0 -> channels 0..7, lane 16 -> 8..15
#pragma unroll
      for (int r = 0; r < 8; ++r) {
        const int ch = r + 8 * half;
        sm.wpm[wave * kCG + ch] = m3r[r];
        sm.wps[wave * kCG + ch] = s3r[r];
        sm.wpv[wave * kCG + ch] = v3r[r];
      }
    }
  }
  __syncthreads();
  if (tid < kCG) {
    float m = -__builtin_inff(), s = 0.f, sv = 0.f;
    for (int wv = 0; wv < kWaves; ++wv) {
      const float mo = sm.wpm[wv * kCG + tid], so = sm.wps[wv * kCG + tid],
                  svo = sm.wpv[wv * kCG + tid];
      const float mn = fmaxf(m, mo);
      s = s * expf(m - mn) + so * expf(mo - mn);
      m = mn;  sv += svo;
    }
    sm.m3[tid] = m; sm.se3[tid] = s; sm.V3[tid] = sv * (1.f / 4096.f);
  }
  __syncthreads();

  // ---- Phase 8: conv3 pass #2 (WMMA) + combine: ss = V2·A3 + V3·A2 --------
  {
    const float mu = sm.scal[0], rs = sm.scal[1];
    for (int t = wave; t < 256; t += kWaves) {
      const int h = t >> 2, w0 = (t & 3) << 4;
      v8f acc = conv3_tile(sm, h, w0, lr, half);
      const int wc = w0 + lr;
      float part = 0.f;
#pragma unroll
      for (int r = 0; r < 8; ++r) {
        const int ch = r + 8 * half;
        const float x3v = fmaxf(acc[r] + sm.b3s[ch], 0.f);
        part += sm.V2[ch] * expf(x3v - sm.m3[ch]) / sm.se3[ch];
        const float x1v = sm.xg[ch * kHW + h * kHWX + wc] *
                          sm.ah[ch * kHWX + h] * sm.aw[ch * kHWX + wc];
        const float x2v = (x1v - mu) * rs * sm.gnw[ch] + sm.gnb[ch];
        part += sm.V3[ch] * expf(x2v - sm.m2[ch]) / sm.se2[ch];
      }
      part += __shfl_xor(part, 16);   // halves hold same position, disjoint channels
      if (half == 0) sm.ss[h * kHWX + wc] = part;
    }
  }
  __syncthreads();

  // ---- Phase 9: softmax over ss (4096) and final gating output ------------
  {
    float mx = -__builtin_inff();
    for (int l = tid; l < kHW; l += kThreads) mx = fmaxf(mx, sm.ss[l]);
    for (int off = 16; off; off >>= 1) mx = fmaxf(mx, __shfl_xor(mx, off));
    if (lane == 0) sm.red[wave] = mx;
  }
  __syncthreads();
  if (tid == 0) {
    float M = -__builtin_inff();
    for (int i = 0; i < kWaves; ++i) M = fmaxf(M, sm.red[i]);
    sm.scal[2] = M;
  }
  __syncthreads();
  {
    const float Ms = sm.scal[2];
    float se = 0.f;
    for (int l = tid; l < kHW; l += kThreads) se += expf(sm.ss[l] - Ms);
    for (int off = 16; off; off >>= 1) se += __shfl_xor(se, off);
    if (lane == 0) sm.red[wave] = se;
  }
  __syncthreads();
  if (tid == 0) {
    float S = 0.f;
    for (int i = 0; i < kWaves; ++i) S += sm.red[i];
    sm.scal[3] = S;
  }
  __syncthreads();
  {
    const float Ms = sm.scal[2];
    const float invS = 1.f / sm.scal[3];
    for (int i = tid * 4; i < kCG * kHW; i += kThreads * 4) {
      const int l = i & 4095;
      float4 o;
      o.x = sm.xg[i + 0] * expf(sm.ss[l + 0] - Ms) * invS;
      o.y = sm.xg[i + 1] * expf(sm.ss[l + 1] - Ms) * invS;
      o.z = sm.xg[i + 2] * expf(sm.ss[l + 2] - Ms) * invS;
      o.w = sm.xg[i + 3] * expf(sm.ss[l + 3] - Ms) * invS;
      *reinterpret_cast<float4*>(ogrp + i) = o;   // GLOBAL_STORE_B128
    }
  }
}

extern "C" void kernel_launch(void* const* d_in, const int* in_sizes, int n_in,
                              void* d_out, int out_size, void* d_ws, size_t ws_size,
                              hipStream_t stream) {
  (void)in_sizes; (void)n_in; (void)out_size; (void)d_ws; (void)ws_size;
  const float* x  = (const float*)d_in[0];
  const float* w1 = (const float*)d_in[1];
  const float* b1 = (const float*)d_in[2];
  const float* wh = (const float*)d_in[3];
  const float* bh = (const float*)d_in[4];
  const float* ww = (const float*)d_in[5];
  const float* bw = (const float*)d_in[6];
  const float* w3 = (const float*)d_in[7];
  const float* b3 = (const float*)d_in[8];
  const float* gw = (const float*)d_in[9];
  const float* gb = (const float*)d_in[10];
  float* out = (float*)d_out;

  static_assert(sizeof(SMem) <= 320 * 1024, "exceeds 320KB WGP LDS");
  // Opt in to >64KB dynamic LDS (idempotent; not a stream op, capture-safe).
  (void)hipFuncSetAttribute((const void*)ema_fused_kernel,
                            hipFuncAttributeMaxDynamicSharedMemorySize,
                            (int)sizeof(SMem));
  ema_fused_kernel<<<kBG, kThreads, sizeof(SMem), stream>>>(
      x, w1, b1, wh, bh, ww, bw, w3, b3, gw, gb, out);
}